// relationLoss_41025527611514
// MI455X (gfx1250) — compile-verified
//
#include <hip/hip_runtime.h>
#include <math.h>
#include <stdint.h>

// ---------------------------------------------------------------------------
// relationLoss: mean over 512 relations of || prod(G[r1]) - prod(G[r2]) ||_F
//   gen_reps : 32 x 256 x 256 fp32,  r1_idx/r2_idx : 512 x 6 int32
// Each block carries a 64x256 fp32 slab through the 6-matrix chain using
// V_WMMA_F32_16X16X4_F32. G panels staged in LDS in WMMA-B fragment order
// (K-quads per column -> conflict-free ds_load_b64 fragments), double-buffered
// and filled with GLOBAL_LOAD_ASYNC_TO_LDS (ASYNCcnt) so the fetch of panel
// p+1 overlaps the WMMA stream on panel p. Final accumulators of both chains
// stay in VGPRs; the Frobenius diff is computed register-vs-register.
// ---------------------------------------------------------------------------

typedef float v2f __attribute__((ext_vector_type(2)));
typedef float v8f __attribute__((ext_vector_type(8)));
typedef int   v4i __attribute__((ext_vector_type(4)));

#define AS1 __attribute__((address_space(1)))
#define AS3 __attribute__((address_space(3)))

#if defined(__has_builtin)
#if __has_builtin(__builtin_amdgcn_global_load_async_to_lds_b32) && \
    __has_builtin(__builtin_amdgcn_global_load_async_to_lds_b128)
#define USE_ASYNC 1
#endif
#endif
#ifndef USE_ASYNC
#define USE_ASYNC 0
#endif

#define NGENS  32
#define DMAT   256
#define NREL   512
#define LCHAIN 6

#define MROWS  64          // rows of chain slab per block
#define CSTR   260         // padded row stride (floats) for C buffers
#define PK     64          // K-panel height staged in LDS
#define GCS    68          // per-column stride (floats) in quad-swizzled G panel
#define NT     256         // 8 waves of 32

#define WB0    0
#define WB1    (MROWS * CSTR)                    // 16640
#define GP0    (2 * MROWS * CSTR)                // 33280
#define GP1    (2 * MROWS * CSTR + DMAT * GCS)   // 50688
#define LDS_FLOATS (2 * MROWS * CSTR + 2 * DMAT * GCS)   // 68096 floats = 272384 B

// Address-space pointer builders (inttoptr: always accepted; AS1 shares the
// flat bit pattern, AS3 is the low-32-bit LDS offset per ISA aperture rules).
__device__ __forceinline__ AS1 int* g32(const float* p)  { return (AS1 int*)(uintptr_t)p; }
__device__ __forceinline__ AS3 int* l32(const float* p)  { return (AS3 int*)(uint32_t)(uintptr_t)p; }
__device__ __forceinline__ AS1 v4i* g128(const float* p) { return (AS1 v4i*)(uintptr_t)p; }
__device__ __forceinline__ AS3 v4i* l128(const float* p) { return (AS3 v4i*)(uint32_t)(uintptr_t)p; }

__device__ __forceinline__ void wait_async0()
{
#if USE_ASYNC
#if __has_builtin(__builtin_amdgcn_s_wait_asynccnt)
    __builtin_amdgcn_s_wait_asynccnt(0);
#else
    asm volatile("s_wait_asynccnt 0" ::: "memory");
#endif
#endif
}

// ---- G panel fill: gp[c*GCS + r] = G[p*PK + r][c]  (r = 0..63, c = 0..255) --
#if USE_ASYNC
// Async: per-lane LDS address does the transpose; no VGPR staging.
// Wave w fills rows [w*8, w*8+8); each instr moves 32 consecutive columns.
__device__ __forceinline__ void fetch_panel(const float* __restrict__ G, float* gp,
                                            int p, int wave, int lane)
{
#pragma unroll 1
    for (int rr = 0; rr < 8; ++rr) {
        const int r = wave * 8 + rr;
        const float* gsrc = G + (p * PK + r) * DMAT + lane;
        float*       ldst = gp + lane * GCS + r;
#pragma unroll
        for (int cg = 0; cg < 8; ++cg) {
            __builtin_amdgcn_global_load_async_to_lds_b32(
                g32(gsrc + cg * 32), l32(ldst + cg * 32 * GCS), 0, 0);
        }
    }
}
// Initial slab rows [r0, r0+64) of G -> C buffer (row-major, padded stride).
__device__ __forceinline__ void fetch_c0(const float* __restrict__ G, float* dst,
                                         int r0, int wave, int lane)
{
#pragma unroll 1
    for (int rr = 0; rr < 8; ++rr) {
        const int r = wave * 8 + rr;
        const float* gsrc = G + (r0 + r) * DMAT + lane * 4;
        float*       ldst = dst + r * CSTR + lane * 4;
        __builtin_amdgcn_global_load_async_to_lds_b128(g128(gsrc), l128(ldst), 0, 0);
        __builtin_amdgcn_global_load_async_to_lds_b128(g128(gsrc + 128), l128(ldst + 128), 0, 0);
    }
}
#else
// Synchronous fallback: 4 coalesced b32 loads per K-quad + aligned b128 store.
__device__ __forceinline__ void fetch_panel(const float* __restrict__ G, float* gp,
                                            int p, int wave, int lane)
{
    const int tid = threadIdx.x;
#pragma unroll
    for (int i = 0; i < 16; ++i) {
        const int q = i * NT + tid;
        const int c = q & 255;
        const int g = q >> 8;
        const float* src = G + (p * PK + g * 4) * DMAT + c;
        float4 v;
        v.x = src[0];
        v.y = src[DMAT];
        v.z = src[2 * DMAT];
        v.w = src[3 * DMAT];
        *(float4*)(gp + c * GCS + g * 4) = v;
    }
}
__device__ __forceinline__ void fetch_c0(const float* __restrict__ G, float* dst,
                                         int r0, int wave, int lane)
{
    const int tid = threadIdx.x;
#pragma unroll
    for (int i = 0; i < 16; ++i) {
        const int f   = i * NT + tid;
        const int row = f >> 6;
        const int cf  = f & 63;
        const float4 v = *(const float4*)(G + (r0 + row) * DMAT + cf * 4);
        *(float4*)(dst + row * CSTR + cf * 4) = v;
    }
}
#endif

// ---- One chain step: C_next(64x256) = C_cur(64x256) * G(256x256) -----------
// Wave w: row tile rt = w&3 (16 rows), column half ch = w>>2 (8 col tiles).
// WRITE_OUT: store result to LDS at dstOff; else keep it in keep[].
template <bool WRITE_OUT>
__device__ __forceinline__ void chain_step(const float* __restrict__ G,
                                           float* lds, int srcOff, int dstOff,
                                           v8f (&keep)[8])
{
    const int tid  = threadIdx.x;
    const int wave = tid >> 5;
    const int lane = tid & 31;
    const int rt   = wave & 3;
    const int ch   = wave >> 2;
    const int hs   = lane >> 4;
    const int l16  = lane & 15;

    const float* Ccur = lds + srcOff;

    v8f acc[8] = {};

    const int arow  = (rt * 16 + l16) * CSTR + hs * 2;   // A-frag per-lane base
    const int bcol  = (ch * 128 + l16) * GCS + hs * 2;   // B-frag per-lane base

    fetch_panel(G, lds + GP0, 0, wave, lane);            // prologue fetch, panel 0

#pragma unroll 1
    for (int p = 0; p < DMAT / PK; ++p) {
        wait_async0();               // own share of panel p landed
        __syncthreads();             // whole panel p visible; prev panel fully consumed
        if (p < DMAT / PK - 1)       // overlap: fetch panel p+1 into other buffer
            fetch_panel(G, lds + (((p + 1) & 1) ? GP1 : GP0), p + 1, wave, lane);

        const float* gp = lds + ((p & 1) ? GP1 : GP0);
        const float* Ca = Ccur + p * PK;

        // K-quad loop, register double-buffered (2 quads in flight)
        v2f a0, a1;
        v2f b0[8], b1[8];
        a0 = *(const v2f*)(Ca + arow);
#pragma unroll
        for (int t = 0; t < 8; ++t)
            b0[t] = *(const v2f*)(gp + bcol + t * 16 * GCS);

#pragma unroll 1
        for (int g = 0; g < PK / 4; g += 2) {
            a1 = *(const v2f*)(Ca + arow + (g + 1) * 4);
#pragma unroll
            for (int t = 0; t < 8; ++t)
                b1[t] = *(const v2f*)(gp + bcol + t * 16 * GCS + (g + 1) * 4);
#pragma unroll
            for (int t = 0; t < 8; ++t)
                acc[t] = __builtin_amdgcn_wmma_f32_16x16x4_f32(
                    false, a0, false, b0[t], (short)0, acc[t], false, false);
            if (g + 2 < PK / 4) {
                a0 = *(const v2f*)(Ca + arow + (g + 2) * 4);
#pragma unroll
                for (int t = 0; t < 8; ++t)
                    b0[t] = *(const v2f*)(gp + bcol + t * 16 * GCS + (g + 2) * 4);
            }
#pragma unroll
            for (int t = 0; t < 8; ++t)
                acc[t] = __builtin_amdgcn_wmma_f32_16x16x4_f32(
                    false, a1, false, b1[t], (short)0, acc[t], false, false);
        }
    }

    if (WRITE_OUT) {
        float* Cnext = lds + dstOff;
        // C/D layout: VGPR v -> rows rt*16+v (lanes 0-15), rt*16+8+v (lanes 16-31)
#pragma unroll
        for (int t = 0; t < 8; ++t) {
            const int n0 = ch * 128 + t * 16;
#pragma unroll
            for (int v = 0; v < 8; ++v) {
                const int row = rt * 16 + hs * 8 + v;
                Cnext[row * CSTR + n0 + l16] = acc[t][v];
            }
        }
    } else {
#pragma unroll
        for (int t = 0; t < 8; ++t) keep[t] = acc[t];
    }
}

// ---- Full chain: final slab stays in accOut registers ----------------------
__device__ __forceinline__ void compute_chain(const float* __restrict__ gens,
                                              const int* __restrict__ idx,
                                              float* lds, int r0, int wave, int lane,
                                              v8f (&accOut)[8])
{
    __syncthreads();   // all waves done with previous chain's LDS
    fetch_c0(gens + (size_t)idx[0] * DMAT * DMAT, lds + WB0, r0, wave, lane);
    int cur = WB0;
#pragma unroll 1
    for (int l = 1; l < LCHAIN - 1; ++l) {
        const int dst = (cur == WB0) ? WB1 : WB0;
        chain_step<true>(gens + (size_t)idx[l] * DMAT * DMAT, lds, cur, dst, accOut);
        cur = dst;
    }
    chain_step<false>(gens + (size_t)idx[LCHAIN - 1] * DMAT * DMAT, lds, cur, 0, accOut);
}

extern "C" __global__ __launch_bounds__(NT)
void relloss_chain_kernel(const float* __restrict__ gens,
                          const int* __restrict__ r1,
                          const int* __restrict__ r2,
                          float* __restrict__ partial)
{
    extern __shared__ float lds[];
    const int rel  = blockIdx.x;
    const int rb   = blockIdx.y;
    const int r0   = rb * MROWS;
    const int tid  = threadIdx.x;
    const int wave = tid >> 5;
    const int lane = tid & 31;

    v8f accA[8], accB[8];
    compute_chain(gens, r1 + rel * LCHAIN, lds, r0, wave, lane, accA);
    compute_chain(gens, r2 + rel * LCHAIN, lds, r0, wave, lane, accB);

    // Frobenius partial, register vs register (tiles are disjoint across waves)
    float s = 0.f;
#pragma unroll
    for (int t = 0; t < 8; ++t)
#pragma unroll
        for (int v = 0; v < 8; ++v) {
            const float d = accA[t][v] - accB[t][v];
            s = fmaf(d, d, s);
        }

    __syncthreads();                 // panel region free for reduction
    float* red = lds + GP0;
    red[tid] = s;
    __syncthreads();
    for (int off = NT / 2; off > 0; off >>= 1) {
        if (tid < off) red[tid] += red[tid + off];
        __syncthreads();
    }
    if (tid == 0) partial[rel * (DMAT / MROWS) + rb] = red[0];
}

extern "C" __global__ __launch_bounds__(NREL)
void relloss_final_kernel(const float* __restrict__ partial, float* __restrict__ out)
{
    __shared__ float red[NREL];
    const int t  = threadIdx.x;
    const int nb = DMAT / MROWS;
    float s = 0.f;
#pragma unroll
    for (int i = 0; i < nb; ++i) s += partial[t * nb + i];
    red[t] = sqrtf(s);
    __syncthreads();
    for (int off = NREL / 2; off > 0; off >>= 1) {
        if (t < off) red[t] += red[t + off];
        __syncthreads();
    }
    if (t == 0) out[0] = red[0] * (1.0f / NREL);
}

extern "C" void kernel_launch(void* const* d_in, const int* in_sizes, int n_in,
                              void* d_out, int out_size, void* d_ws, size_t ws_size,
                              hipStream_t stream)
{
    const float* gens = (const float*)d_in[0];
    const int*   r1   = (const int*)d_in[1];
    const int*   r2   = (const int*)d_in[2];
    float* partial    = (float*)d_ws;            // 512 * 4 floats, fully overwritten
    float* out        = (float*)d_out;

    const size_t ldsBytes = (size_t)LDS_FLOATS * sizeof(float);  // ~266 KB of 320 KB
    dim3 grid(NREL, DMAT / MROWS);
    relloss_chain_kernel<<<grid, NT, ldsBytes, stream>>>(gens, r1, r2, partial);
    relloss_final_kernel<<<1, NREL, 0, stream>>>(partial, out);
}